// SatelliteEvolveGCN_9088150799041
// MI455X (gfx1250) — compile-verified
//
#include <hip/hip_runtime.h>
#include <math.h>

// ---------------------------------------------------------------------------
// SatelliteEvolveGCN, MI455X (gfx1250), wave32.
// Only the final timestep's GCN conv reaches the output (scan overwrites h),
// and global-mean-pool collapses the scatter-add into a 3-float sum of all
// edge messages.  Memory-bound (~30 MB HBM traffic ≈ 1.3 us at 23.3 TB/s);
// the one dense GEMM (x[T-1] @ w_final) runs on V_WMMA_F32_16X16X4_F32.
// ---------------------------------------------------------------------------

typedef __attribute__((ext_vector_type(2))) float v2f;
typedef __attribute__((ext_vector_type(8))) float v8f;

#define T_STEPS 20
#define N_NODES 100000
#define E_EDGES 1600000

#define BLK 256
#define GB_EDGE 1024   // blocks for edge-message reduction (grid-stride)
#define GB_NODE 256    // blocks for self-loop / dinv pass (grid-stride)

#define XW_BLK 64      // 2 waves/block; 3125 blocks * 2 waves * 16 rows = N

// ---------------------------------------------------------------------------
// 1) GRU weight evolution: 3x3 state, 20 sequential steps.  Scalar, 1 thread.
//    PyTorch gate order r,z,n:  gi = w @ Wi^T + bi,  gh = w @ Wh^T + bh
// ---------------------------------------------------------------------------
__global__ void gru_evolve_kernel(const float* __restrict__ W0,
                                  const float* __restrict__ Wi,   // [9,3]
                                  const float* __restrict__ Wh,   // [9,3]
                                  const float* __restrict__ bi,   // [9]
                                  const float* __restrict__ bh,   // [9]
                                  float* __restrict__ w_out) {    // [9]
  if (threadIdx.x != 0 || blockIdx.x != 0) return;
  float w[9];
#pragma unroll
  for (int i = 0; i < 9; ++i) w[i] = W0[i];
  for (int t = 0; t < T_STEPS; ++t) {
    float wn[9];
    for (int i = 0; i < 3; ++i) {            // weight row used as batch row
      float gi[9], gh[9];
#pragma unroll
      for (int g = 0; g < 9; ++g) {
        float si = bi[g], sh = bh[g];
#pragma unroll
        for (int k = 0; k < 3; ++k) {
          si += w[i * 3 + k] * Wi[g * 3 + k];
          sh += w[i * 3 + k] * Wh[g * 3 + k];
        }
        gi[g] = si; gh[g] = sh;
      }
#pragma unroll
      for (int j = 0; j < 3; ++j) {
        float r = 1.0f / (1.0f + expf(-(gi[j]     + gh[j])));
        float z = 1.0f / (1.0f + expf(-(gi[3 + j] + gh[3 + j])));
        float n = tanhf(gi[6 + j] + r * gh[6 + j]);
        wn[i * 3 + j] = (1.0f - z) * n + z * w[i * 3 + j];
      }
    }
#pragma unroll
    for (int i = 0; i < 9; ++i) w[i] = wn[i];
  }
#pragma unroll
  for (int i = 0; i < 9; ++i) w_out[i] = w[i];
}

// ---------------------------------------------------------------------------
// 2) Degree: deg[n] = 1 (self loop) + #edges with dst == n.
//    Integer atomics are exact => deterministic.  b128 streaming reads.
// ---------------------------------------------------------------------------
__global__ void deg_init_kernel(unsigned* __restrict__ deg) {
  int n = blockIdx.x * blockDim.x + threadIdx.x;
  if (n < N_NODES) deg[n] = 1u;
}

__global__ __launch_bounds__(BLK) void deg_hist_kernel(
    const int4* __restrict__ dst4, unsigned* __restrict__ deg) {
  const int nq = E_EDGES / 4;
  for (int i = blockIdx.x * blockDim.x + threadIdx.x; i < nq;
       i += gridDim.x * blockDim.x) {
    int4 d = dst4[i];
    atomicAdd(&deg[d.x], 1u);
    atomicAdd(&deg[d.y], 1u);
    atomicAdd(&deg[d.z], 1u);
    atomicAdd(&deg[d.w], 1u);
  }
}

// ---------------------------------------------------------------------------
// 3) xw = x[T-1] @ w_final via V_WMMA_F32_16X16X4_F32.
//    One wave -> one 16-row tile; K padded 3->4, B = w zero-padded to 4x16.
//    N divisible by 16 and grid covers exactly N/16 waves => no bounds checks.
//    A (16x4 f32): lanes 0-15 hold K=0,1 in v[0],v[1]; lanes 16-31 hold K=2,3.
//    B (4x16 f32): lane-half 0 holds rows K=0,1; lane-half 1 holds K=2,3.
//    D (16x16):    VGPR r = row r (lanes 0-15) and row r+8 (lanes 16-31).
// ---------------------------------------------------------------------------
__global__ __launch_bounds__(XW_BLK) void xw_wmma_kernel(
    const float* __restrict__ x,      // [N,3] last timestep
    const float* __restrict__ w,      // [3,3] evolved weight
    float* __restrict__ xw) {         // [N,3]
  const int lane = threadIdx.x & 31;
  const int half = lane >> 4;
  const int lh   = lane & 15;
  const int wave = blockIdx.x * (XW_BLK / 32) + (threadIdx.x >> 5);
  const int m0   = wave * 16;
  const int row  = m0 + lh;

  v2f a;
  if (half == 0) { a.x = x[row * 3 + 0]; a.y = x[row * 3 + 1]; }
  else           { a.x = x[row * 3 + 2]; a.y = 0.0f; /* K=3 pad */ }

  v2f b; b.x = 0.0f; b.y = 0.0f;
  if (lh < 3) {
    if (half == 0) { b.x = w[0 * 3 + lh]; b.y = w[1 * 3 + lh]; }
    else           { b.x = w[2 * 3 + lh]; /* K=3 pad */ }
  }

  v8f c = {};
  // (neg_a, A, neg_b, B, c_mod, C, reuse_a, reuse_b) -- all lanes execute.
  c = __builtin_amdgcn_wmma_f32_16x16x4_f32(false, a, false, b, (short)0, c,
                                            false, false);

  if (lh < 3) {
#pragma unroll
    for (int r = 0; r < 8; ++r)
      xw[(m0 + r + 8 * half) * 3 + lh] = c[r];
  }
}

// ---------------------------------------------------------------------------
// deterministic 3-channel block reduction (fixed-shape LDS tree)
// ---------------------------------------------------------------------------
__device__ inline void block_reduce3_store(float s0, float s1, float s2,
                                           float* __restrict__ out3) {
  __shared__ float r0[BLK], r1[BLK], r2[BLK];
  r0[threadIdx.x] = s0; r1[threadIdx.x] = s1; r2[threadIdx.x] = s2;
  __syncthreads();
  for (int off = BLK >> 1; off > 0; off >>= 1) {
    if ((int)threadIdx.x < off) {
      r0[threadIdx.x] += r0[threadIdx.x + off];
      r1[threadIdx.x] += r1[threadIdx.x + off];
      r2[threadIdx.x] += r2[threadIdx.x + off];
    }
    __syncthreads();
  }
  if (threadIdx.x == 0) { out3[0] = r0[0]; out3[1] = r1[0]; out3[2] = r2[0]; }
}

// ---------------------------------------------------------------------------
// 4) dinv[n] = 1/sqrt(deg[n]) (in-place over deg) + self-loop messages:
//    partial += xw[n] * dinv[n]^2
// ---------------------------------------------------------------------------
__global__ __launch_bounds__(BLK) void self_kernel(
    const float* __restrict__ xw, unsigned* __restrict__ deg_dinv,
    float* __restrict__ partials) {
  float s0 = 0.f, s1 = 0.f, s2 = 0.f;
  for (int n = blockIdx.x * blockDim.x + threadIdx.x; n < N_NODES;
       n += gridDim.x * blockDim.x) {
    float di = 1.0f / sqrtf((float)deg_dinv[n]);
    ((float*)deg_dinv)[n] = di;           // each n owned by exactly one thread
    float nn = di * di;
    s0 += xw[n * 3 + 0] * nn;
    s1 += xw[n * 3 + 1] * nn;
    s2 += xw[n * 3 + 2] * nn;
  }
  block_reduce3_store(s0, s1, s2, &partials[(GB_EDGE + blockIdx.x) * 3]);
}

// ---------------------------------------------------------------------------
// 5) Edge messages: partial += xw[src] * dinv[src] * dinv[dst]
//    b128 index streaming; gather tables L2-resident (dinv 0.4MB, xw 1.2MB).
// ---------------------------------------------------------------------------
__global__ __launch_bounds__(BLK) void edge_kernel(
    const int4* __restrict__ src4, const int4* __restrict__ dst4,
    const float* __restrict__ xw, const float* __restrict__ dinv,
    float* __restrict__ partials) {
  float s0 = 0.f, s1 = 0.f, s2 = 0.f;
  const int nq = E_EDGES / 4;
  for (int i = blockIdx.x * blockDim.x + threadIdx.x; i < nq;
       i += gridDim.x * blockDim.x) {
    int4 s = src4[i];
    int4 d = dst4[i];
#pragma unroll
    for (int k = 0; k < 4; ++k) {
      int ss = (k == 0) ? s.x : (k == 1) ? s.y : (k == 2) ? s.z : s.w;
      int dd = (k == 0) ? d.x : (k == 1) ? d.y : (k == 2) ? d.z : d.w;
      float nm = dinv[ss] * dinv[dd];
      s0 += xw[ss * 3 + 0] * nm;
      s1 += xw[ss * 3 + 1] * nm;
      s2 += xw[ss * 3 + 2] * nm;
    }
  }
  block_reduce3_store(s0, s1, s2, &partials[blockIdx.x * 3]);
}

// ---------------------------------------------------------------------------
// 6) Final: ordered sum of block partials, mean-pool, linear head.
// ---------------------------------------------------------------------------
__global__ void final_kernel(const float* __restrict__ partials,
                             const float* __restrict__ conv_bias, // [3]
                             const float* __restrict__ lin_W,     // [2,3]
                             const float* __restrict__ lin_b,     // [2]
                             float* __restrict__ out) {           // [2]
  if (threadIdx.x != 0 || blockIdx.x != 0) return;
  float g0 = 0.f, g1 = 0.f, g2 = 0.f;
  for (int b = 0; b < GB_EDGE + GB_NODE; ++b) {
    g0 += partials[b * 3 + 0];
    g1 += partials[b * 3 + 1];
    g2 += partials[b * 3 + 2];
  }
  const float invN = 1.0f / (float)N_NODES;
  g0 = g0 * invN + conv_bias[0];
  g1 = g1 * invN + conv_bias[1];
  g2 = g2 * invN + conv_bias[2];
#pragma unroll
  for (int j = 0; j < 2; ++j)
    out[j] = g0 * lin_W[j * 3 + 0] + g1 * lin_W[j * 3 + 1] +
             g2 * lin_W[j * 3 + 2] + lin_b[j];
}

// ---------------------------------------------------------------------------
extern "C" void kernel_launch(void* const* d_in, const int* in_sizes, int n_in,
                              void* d_out, int out_size, void* d_ws,
                              size_t ws_size, hipStream_t stream) {
  const float* x      = (const float*)d_in[0];  // [T,N,3]
  const int*   ei     = (const int*)d_in[1];    // [T,2,E]
  const float* W0     = (const float*)d_in[2];  // [3,3]
  const float* gru_Wi = (const float*)d_in[3];  // [9,3]
  const float* gru_Wh = (const float*)d_in[4];  // [9,3]
  const float* gru_bi = (const float*)d_in[5];  // [9]
  const float* gru_bh = (const float*)d_in[6];  // [9]
  const float* cbias  = (const float*)d_in[7];  // [3]
  const float* lin_W  = (const float*)d_in[8];  // [2,3]
  const float* lin_b  = (const float*)d_in[9];  // [2]
  float* out = (float*)d_out;                   // [2]

  // Only the last timestep feeds the output.
  const float* x_last = x + (size_t)(T_STEPS - 1) * N_NODES * 3;
  const int* src = ei + (size_t)(T_STEPS - 1) * 2 * E_EDGES;
  const int* dst = src + E_EDGES;               // 16B-aligned (E%4==0)

  // Workspace carve-out (base is 256B-aligned from hipMalloc).
  float*    w_final  = (float*)d_ws;                          // 16 floats
  float*    xw       = w_final + 16;                          // N*3 floats
  unsigned* deg      = (unsigned*)(xw + (size_t)N_NODES * 3); // N u32 -> f32
  float*    partials = (float*)(deg + N_NODES);  // (GB_EDGE+GB_NODE)*3 floats

  gru_evolve_kernel<<<1, 1, 0, stream>>>(W0, gru_Wi, gru_Wh, gru_bi, gru_bh,
                                         w_final);

  deg_init_kernel<<<(N_NODES + BLK - 1) / BLK, BLK, 0, stream>>>(deg);
  deg_hist_kernel<<<GB_EDGE, BLK, 0, stream>>>((const int4*)dst, deg);

  const int xw_blocks = N_NODES / (16 * (XW_BLK / 32));  // 3125, exact
  xw_wmma_kernel<<<xw_blocks, XW_BLK, 0, stream>>>(x_last, w_final, xw);

  self_kernel<<<GB_NODE, BLK, 0, stream>>>(xw, deg, partials);
  edge_kernel<<<GB_EDGE, BLK, 0, stream>>>((const int4*)src, (const int4*)dst,
                                           xw, (const float*)deg, partials);

  final_kernel<<<1, 1, 0, stream>>>(partials, cbias, lin_W, lin_b, out);
}